// SE3_Transformer_40759239639121
// MI455X (gfx1250) — compile-verified
//
#include <hip/hip_runtime.h>
#include <math.h>

#define N_NODES 8192
#define N_EDGES 65536

typedef float v8f __attribute__((ext_vector_type(8)));
typedef float v2f __attribute__((ext_vector_type(2)));

// ---------------- workspace layout (floats) ----------------
#define OFF_Z    0                       // 8192
#define OFF_EXP  8192                    // 65536
#define OFF_M0   (8192 + 65536)          // 256
#define OFF_M1   (OFF_M0 + 256)          // 256
#define OFF_QD   (OFF_M1 + 256)          // 8192*64
#define OFF_V    (OFF_QD + N_NODES * 64) // 65536*64

__device__ __forceinline__ float sus(float x) {
  // soft_unit_step
  return x > 0.f ? __expf(-1.f / x) : 0.f;
}

__device__ __forceinline__ v8f wmma_step(v2f a, v2f b, v8f c) {
  // D = A(16x4,f32) x B(4x16,f32) + C
  return __builtin_amdgcn_wmma_f32_16x16x4_f32(false, a, false, b, (short)0, c,
                                               false, false);
}

// ---------------- zero out + z ----------------
__global__ void zero_kernel(float* __restrict__ out, float* __restrict__ z) {
  int i = blockIdx.x * blockDim.x + threadIdx.x;
  if (i < N_NODES * 64) out[i] = 0.f;
  if (i < N_NODES) z[i] = 0.f;
}

// ---------------- fold Wq/Wd into single matrices ----------------
__global__ void fold_kernel(const float* __restrict__ Wq0, const float* __restrict__ Wq1,
                            const float* __restrict__ Wd0, const float* __restrict__ Wd1,
                            float* __restrict__ M0, float* __restrict__ M1) {
  int i = threadIdx.x;            // 256 threads, one (row,col) each
  int r = i >> 4, c = i & 15;
  float ad = rsqrtf(512.f);       // 1/sqrt(2*MUL*MUL)
  float s0 = ad * 0.25f;          // * 1/sqrt(MUL)
  float s1 = s0 * 0.57735026919f; // extra 1/sqrt(3)
  float a0 = 0.f, a1 = 0.f;
  for (int t = 0; t < 16; ++t) {
    a0 += Wq0[r * 16 + t] * Wd0[t * 16 + c];
    a1 += Wq1[r * 16 + t] * Wd1[t * 16 + c];
  }
  M0[i] = a0 * s0;
  M1[i] = a1 * s1;
}

// ---------------- node queries: qd = f @ M (WMMA) ----------------
__global__ void __launch_bounds__(256, 1)
node_q_kernel(const float* __restrict__ f_in, const float* __restrict__ M0,
              const float* __restrict__ M1, float* __restrict__ qd) {
  int lane = threadIdx.x & 31;
  int n = lane & 15, h = lane >> 4;
  int tile = blockIdx.x * 8 + (threadIdx.x >> 5); // 512 tiles of 16 nodes

  v2f b0[4], b1[4], a0[4];
#pragma unroll
  for (int c = 0; c < 4; ++c) {
    int k0 = 4 * c + 2 * h;
    b0[c][0] = M0[k0 * 16 + n];
    b0[c][1] = M0[(k0 + 1) * 16 + n];
    b1[c][0] = M1[k0 * 16 + n];
    b1[c][1] = M1[(k0 + 1) * 16 + n];
    a0[c] = *(const v2f*)(f_in + (tile * 16 + n) * 64 + k0); // f0 row, A layout
  }
  v8f d = {0.f, 0.f, 0.f, 0.f, 0.f, 0.f, 0.f, 0.f};
#pragma unroll
  for (int c = 0; c < 4; ++c) d = wmma_step(a0[c], b0[c], d);
#pragma unroll
  for (int r = 0; r < 8; ++r)
    qd[(tile * 16 + r + 8 * h) * 64 + n] = d[r];

#pragma unroll
  for (int cc = 0; cc < 3; ++cc) {
    v2f a1[4];
#pragma unroll
    for (int c = 0; c < 4; ++c) {
      int k0 = 4 * c + 2 * h;
      const float* base = f_in + (tile * 16 + n) * 64 + 16;
      a1[c][0] = base[k0 * 3 + cc];
      a1[c][1] = base[(k0 + 1) * 3 + cc];
    }
    v8f d1 = {0.f, 0.f, 0.f, 0.f, 0.f, 0.f, 0.f, 0.f};
#pragma unroll
    for (int c = 0; c < 4; ++c) d1 = wmma_step(a1[c], b1[c], d1);
#pragma unroll
    for (int r = 0; r < 8; ++r)
      qd[(tile * 16 + r + 8 * h) * 64 + 16 + n * 3 + cc] = d1[r];
  }
}

// ---------------- fused per-edge kernel ----------------
// 8 waves/block, one 16-edge tile per wave. Wk2/Wv2 pre-swizzled into LDS.
__global__ void __launch_bounds__(256, 1)
edge_kernel(const float* __restrict__ f_in, const float* __restrict__ edge_vec,
            const int* __restrict__ edge_src, const int* __restrict__ edge_dst,
            const float* __restrict__ Wk1, const float* __restrict__ Wv1,
            const float* __restrict__ Wk2, const float* __restrict__ Wv2,
            const float* __restrict__ qd, float* __restrict__ zbuf,
            float* __restrict__ expbuf, float* __restrict__ vws) {
  extern __shared__ float smem[];
  const int tid = threadIdx.x;
  const int lane = tid & 31;
  const int wid = tid >> 5;
  const int n = lane & 15;
  const int h = lane >> 4;
  const float W2SCALE = 0.0441941738f; // (1/sqrt(HID)) * (1/sqrt(2*MUL))

  // Preload W2 (k then v), rearranged into per-lane WMMA-B consumption order:
  // float idx = (((m*64+nt)*4 + c)*32 + h*16 + n)*2 + (t&1), value W2[t][nt*16+n]
  for (int idx = tid; idx < 2 * 16 * 1024; idx += 256) {
    int m = idx >> 14;
    int rem = idx & 16383;
    int t = rem >> 10;
    int col = rem & 1023;
    int nt = col >> 4, nn = col & 15;
    int c = t >> 2, hh = (t >> 1) & 1, p = t & 1;
    const float* W2 = m ? Wv2 : Wk2;
    smem[((((m * 64 + nt) * 4 + c) * 32 + hh * 16 + nn) << 1) + p] =
        W2[t * 1024 + col] * W2SCALE;
  }

  // per-wave scratch: [0,512) H(k,v) | [512,768) x0 | [768,1024) xs |
  //                   [1024,1792) x1 | [1792,1920) edge records (16 x 8)
  float* pw = smem + 32768 + wid * 1920;
  const int tileBase = (blockIdx.x * 8 + wid) * 16;

  if (lane < 16) {
    int ge = tileBase + lane;
    float vx = edge_vec[ge * 3 + 0], vy = edge_vec[ge * 3 + 1], vz = edge_vec[ge * 3 + 2];
    float rr = fmaxf(sqrtf(vx * vx + vy * vy + vz * vz), 1e-12f);
    float inv = 1.7320508076f / rr; // sh1 = sqrt(3)*u
    float* rec = pw + 1792 + lane * 8;
    rec[0] = vx * inv;
    rec[1] = vy * inv;
    rec[2] = vz * inv;
    rec[3] = sus(10.f * (1.f - rr * 0.2f)); // edge_cut
    rec[4] = rr;
    rec[5] = __int_as_float(edge_src[ge]);
    rec[6] = __int_as_float(edge_dst[ge]);
    rec[7] = 0.f;
  }
  __syncthreads();

  // gather x0 / xs = (x1 . sh)/sqrt(3) / raw x1 per (edge,i)
  for (int p = lane; p < 256; p += 32) {
    int e = p >> 4, i = p & 15;
    const float* rec = pw + 1792 + e * 8;
    int src = __float_as_int(rec[5]);
    const float* fr = f_in + src * 64;
    float x0v = fr[i];
    float xa = fr[16 + i * 3 + 0], xb = fr[16 + i * 3 + 1], xc = fr[16 + i * 3 + 2];
    pw[512 + p] = x0v;
    pw[768 + p] = (xa * rec[0] + xb * rec[1] + xc * rec[2]) * 0.5773502692f;
    pw[1024 + p * 3 + 0] = xa;
    pw[1024 + p * 3 + 1] = xb;
    pw[1024 + p * 3 + 2] = xc;
  }

  // ---- stage 1: H = silu(rb @ W1/sqrt(NB)) for k and v (WMMA) ----
  float rM = pw[1792 + n * 8 + 4]; // r of edge m=n (written before last sync)
  float tpos = rM * 3.4f;          // r/step, step = 5/17
  v2f aR[4], bK[4], bV[4];
#pragma unroll
  for (int c = 0; c < 4; ++c) {
    int k0 = 4 * c + 2 * h;
    float b0f = (float)k0, b1f = (float)(k0 + 1);
    aR[c][0] = 33.7342922f * sus(tpos - b0f) * sus(b0f + 2.f - tpos);
    aR[c][1] = 33.7342922f * sus(tpos - b1f) * sus(b1f + 2.f - tpos);
    bK[c][0] = Wk1[k0 * 16 + n] * 0.25f;
    bK[c][1] = Wk1[(k0 + 1) * 16 + n] * 0.25f;
    bV[c][0] = Wv1[k0 * 16 + n] * 0.25f;
    bV[c][1] = Wv1[(k0 + 1) * 16 + n] * 0.25f;
  }
  v8f dK = {0.f, 0.f, 0.f, 0.f, 0.f, 0.f, 0.f, 0.f};
  v8f dV = {0.f, 0.f, 0.f, 0.f, 0.f, 0.f, 0.f, 0.f};
#pragma unroll
  for (int c = 0; c < 4; ++c) {
    dK = wmma_step(aR[c], bK[c], dK);
    dV = wmma_step(aR[c], bV[c], dV);
  }
#pragma unroll
  for (int r = 0; r < 8; ++r) { // silu, store D-layout -> LDS
    float tk = dK[r], tv = dV[r];
    pw[(r + 8 * h) * 16 + n] = tk / (1.f + __expf(-tk));
    pw[256 + (r + 8 * h) * 16 + n] = tv / (1.f + __expf(-tv));
  }
  __syncthreads();

  v2f hA[2][4]; // reload H in WMMA-A layout
#pragma unroll
  for (int c = 0; c < 4; ++c) {
    hA[0][c] = *(const v2f*)(pw + n * 16 + 4 * c + 2 * h);
    hA[1][c] = *(const v2f*)(pw + 256 + n * 16 + 4 * c + 2 * h);
  }

  // ---- stage 2: 64 N-tiles of wk/wv, consumed immediately by fctp ----
  float a0[2][8], a2[2][8], a3[2][3][8];
#pragma unroll
  for (int r = 0; r < 8; ++r) {
    a0[0][r] = 0.f; a0[1][r] = 0.f;
    a2[0][r] = 0.f; a2[1][r] = 0.f;
    a3[0][0][r] = 0.f; a3[0][1][r] = 0.f; a3[0][2][r] = 0.f;
    a3[1][0][r] = 0.f; a3[1][1][r] = 0.f; a3[1][2][r] = 0.f;
  }
#pragma unroll 1
  for (int nt = 0; nt < 64; ++nt) {
    int s = nt >> 4, i = nt & 15;
    v8f dk = {0.f, 0.f, 0.f, 0.f, 0.f, 0.f, 0.f, 0.f};
    v8f dv = {0.f, 0.f, 0.f, 0.f, 0.f, 0.f, 0.f, 0.f};
#pragma unroll
    for (int c = 0; c < 4; ++c) {
      v2f bk = *(const v2f*)(smem + (((nt * 4 + c) * 32 + lane) << 1));
      v2f bv = *(const v2f*)(smem + ((((64 + nt) * 4 + c) * 32 + lane) << 1));
      dk = wmma_step(hA[0][c], bk, dk);
      dv = wmma_step(hA[1][c], bv, dv);
    }
    if (s == 3) {
#pragma unroll
      for (int r = 0; r < 8; ++r) {
        const float* x1p = pw + 1024 + ((r + 8 * h) * 16 + i) * 3;
        float xk = dk[r], xv = dv[r];
        a3[0][0][r] += x1p[0] * xk; a3[1][0][r] += x1p[0] * xv;
        a3[0][1][r] += x1p[1] * xk; a3[1][1][r] += x1p[1] * xv;
        a3[0][2][r] += x1p[2] * xk; a3[1][2][r] += x1p[2] * xv;
      }
    } else {
      int off = (s == 1) ? 768 : 512; // xs for slice 1, x0 for slices 0 & 2
#pragma unroll
      for (int r = 0; r < 8; ++r) {
        float x = pw[off + (r + 8 * h) * 16 + i];
        if (s == 2) { a2[0][r] += x * dk[r]; a2[1][r] += x * dv[r]; }
        else        { a0[0][r] += x * dk[r]; a0[1][r] += x * dv[r]; }
      }
    }
  }

  // ---- epilogue: dot, exp, z-atomic, store v ----
#pragma unroll
  for (int r = 0; r < 8; ++r) {
    int e = r + 8 * h;
    const float* rec = pw + 1792 + e * 8;
    int dste = __float_as_int(rec[6]);
    float sh0 = rec[0], sh1 = rec[1], sh2 = rec[2];
    const float* qrow = qd + dste * 64;
    float t2 = a2[0][r];
    float p = qrow[n] * a0[0][r]
            + qrow[16 + n * 3 + 0] * (t2 * sh0 + a3[0][0][r])
            + qrow[16 + n * 3 + 1] * (t2 * sh1 + a3[0][1][r])
            + qrow[16 + n * 3 + 2] * (t2 * sh2 + a3[0][2][r]);
    p += __shfl_xor(p, 1);
    p += __shfl_xor(p, 2);
    p += __shfl_xor(p, 4);
    p += __shfl_xor(p, 8); // sum over j within each 16-lane half
    int ge = tileBase + e;
    if (n == r) {
      float ex = rec[3] * __expf(p);
      expbuf[ge] = ex;
      atomicAdd(zbuf + dste, ex);
    }
    float* vrow = vws + ge * 64;
    float tv = a2[1][r];
    vrow[n] = a0[1][r];
    vrow[16 + n * 3 + 0] = tv * sh0 + a3[1][0][r];
    vrow[16 + n * 3 + 1] = tv * sh1 + a3[1][1][r];
    vrow[16 + n * 3 + 2] = tv * sh2 + a3[1][2][r];
  }
}

// ---------------- finalize: alpha, sqrt, scatter-add ----------------
__global__ void finalize_kernel(const int* __restrict__ edge_dst,
                                const float* __restrict__ expbuf,
                                const float* __restrict__ z,
                                const float* __restrict__ vws,
                                float* __restrict__ out) {
  int t = blockIdx.x * blockDim.x + threadIdx.x;
  if (t >= N_EDGES * 16) return;
  int e = t >> 4, j = t & 15;
  int d = edge_dst[e];
  float zz = z[d];
  zz = (zz == 0.f) ? 1.f : zz;
  float alpha = expbuf[e] / zz;
  float s = alpha > 0.f ? sqrtf(alpha) : 0.f;
  const float* vrow = vws + e * 64;
  float* orow = out + d * 64;
  atomicAdd(orow + j, s * vrow[j]);
  atomicAdd(orow + 16 + j * 3 + 0, s * vrow[16 + j * 3 + 0]);
  atomicAdd(orow + 16 + j * 3 + 1, s * vrow[16 + j * 3 + 1]);
  atomicAdd(orow + 16 + j * 3 + 2, s * vrow[16 + j * 3 + 2]);
}

extern "C" void kernel_launch(void* const* d_in, const int* in_sizes, int n_in,
                              void* d_out, int out_size, void* d_ws, size_t ws_size,
                              hipStream_t stream) {
  const float* f_in     = (const float*)d_in[0];
  const float* edge_vec = (const float*)d_in[1];
  const int*   edge_src = (const int*)d_in[2];
  const int*   edge_dst = (const int*)d_in[3];
  const float* Wq0 = (const float*)d_in[4];
  const float* Wq1 = (const float*)d_in[5];
  const float* Wk1 = (const float*)d_in[6];
  const float* Wk2 = (const float*)d_in[7];
  const float* Wv1 = (const float*)d_in[8];
  const float* Wv2 = (const float*)d_in[9];
  const float* Wd0 = (const float*)d_in[10];
  const float* Wd1 = (const float*)d_in[11];
  float* out = (float*)d_out;
  float* ws  = (float*)d_ws;

  float* zbuf   = ws + OFF_Z;
  float* expbuf = ws + OFF_EXP;
  float* M0     = ws + OFF_M0;
  float* M1     = ws + OFF_M1;
  float* qdbuf  = ws + OFF_QD;
  float* vbuf   = ws + OFF_V;

  size_t smemB = (size_t)(32768 + 8 * 1920) * sizeof(float); // 192512 B < 320KB WGP LDS
  hipFuncSetAttribute((const void*)edge_kernel,
                      hipFuncAttributeMaxDynamicSharedMemorySize, (int)smemB);

  zero_kernel<<<(N_NODES * 64) / 256, 256, 0, stream>>>(out, zbuf);
  fold_kernel<<<1, 256, 0, stream>>>(Wq0, Wq1, Wd0, Wd1, M0, M1);
  node_q_kernel<<<N_NODES / 128, 256, 0, stream>>>(f_in, M0, M1, qdbuf);
  edge_kernel<<<N_EDGES / 128, 256, smemB, stream>>>(f_in, edge_vec, edge_src, edge_dst,
                                                     Wk1, Wv1, Wk2, Wv2, qdbuf, zbuf,
                                                     expbuf, vbuf);
  finalize_kernel<<<(N_EDGES * 16) / 256, 256, 0, stream>>>(edge_dst, expbuf, zbuf,
                                                            vbuf, out);
}